// SimpleConvolutionNet_11424613007851
// MI455X (gfx1250) — compile-verified
//
#include <hip/hip_runtime.h>

// z = segment_sum(node[i1] @ Wn.T + bn, i0)   -- softmax over [E,1] axis==1 is 1.0,
// so the entire h0 / edge-MLP / attention path in the reference is dead code.

typedef __attribute__((ext_vector_type(2))) float v2f;
typedef __attribute__((ext_vector_type(8))) float v8f;

#define IN_C  64
#define OUT_C 32

__global__ __launch_bounds__(256) void gat_zero_kernel(float* __restrict__ out, int n) {
    int i = blockIdx.x * blockDim.x + threadIdx.x;
    if (i < n) out[i] = 0.0f;
}

// Scalar fallback for edges not covered by full 16-edge WMMA tiles (E % 16).
__global__ void gat_edge_tail_kernel(const float* __restrict__ node,
                                     const int* __restrict__ idx,
                                     const float* __restrict__ Wn,
                                     const float* __restrict__ bn,
                                     float* __restrict__ out,
                                     int E, int estart) {
    int t = blockIdx.x * blockDim.x + threadIdx.x;
    int e = estart + (t >> 5);
    int n = t & 31;
    if (e >= E) return;
    unsigned i0 = (unsigned)idx[e];
    unsigned i1 = (unsigned)idx[E + e];
    const float* row = node + i1 * (unsigned)IN_C;
    const float* w   = Wn   + (unsigned)n * IN_C;
    float acc = bn[n];
    #pragma unroll
    for (int k = 0; k < IN_C; ++k) acc += row[k] * w[k];
    unsafeAtomicAdd(out + i0 * (unsigned)OUT_C + n, acc);
}

// Persistent wave32 kernel. Each wave:
//   - preloads its B-operand slice of Wn.T into 64 VGPRs ONCE (per-lane B elements
//     are contiguous pairs of the row-major Wn, so plain aligned v2f global loads
//     land directly in the even-aligned register pairs WMMA needs -> no LDS, no movs)
//   - grid-strides over 16-edge tiles: gather 16 node rows (16x64 f32), run
//     16 k-steps x 2 N-tiles of V_WMMA_F32_16X16X4_F32, add bias, and
//     scatter-add the 16x32 h1 tile into out[i0] with native f32 atomics.
// All hot-loop offsets are unsigned 32-bit so the compiler can use the GVS
// addressing mode (SGPR64 base + 32-bit VGPR offset) instead of 64-bit VALU adds.
__global__ __launch_bounds__(256) void gat_edge_wmma_kernel(const float* __restrict__ node,
                                                            const int* __restrict__ idx,
                                                            const float* __restrict__ Wn,
                                                            const float* __restrict__ bn,
                                                            float* __restrict__ out,
                                                            int E, int numTiles,
                                                            int totalWaves) {
    int tid    = threadIdx.x;
    int lane   = tid & 31;
    unsigned m = lane & 15;                 // output column within N-tile / edge row id
    unsigned half = lane >> 4;              // 0: K pair {0,1}, 1: K pair {2,3}
    int waveId = blockIdx.x * (blockDim.x >> 5) + (tid >> 5);

    // B (4x16 f32 per k-step), mirrored A layout: lane holds B[2*half][n], B[2*half+1][n]
    // with n = lane&15. B[k][n] = Wn[n*64 + k] -> contiguous pair at Wn + n*64 + 4s + 2*half.
    v2f B0[16], B1[16];
    const float* w0 = Wn + m * IN_C + 2u * half;            // N-tile 0: n = m
    const float* w1 = Wn + (16u + m) * IN_C + 2u * half;    // N-tile 1: n = m + 16
    #pragma unroll
    for (int s = 0; s < 16; ++s) {
        B0[s] = *(const v2f*)(w0 + 4 * s);
        B1[s] = *(const v2f*)(w1 + 4 * s);
    }
    float bn0 = bn[m];
    float bn1 = bn[16u + m];

    for (int tile = waveId; tile < numTiles; tile += totalWaves) {
        unsigned e0  = (unsigned)tile * 16u;
        unsigned src = (unsigned)idx[E + e0 + m];   // i1 (gather source)
        unsigned dst = (unsigned)idx[e0 + m];       // i0 (scatter destination)

        // A (16x4 f32 per k-step): lanes 0-15 hold K=k0,k0+1 of row m; lanes 16-31 K+2,K+3.
        unsigned aoff = src * (unsigned)IN_C + 2u * half;   // 32-bit element offset

        v8f c0 = {};                        // N = 0..15
        v8f c1 = {};                        // N = 16..31
        #pragma unroll
        for (int s = 0; s < 16; ++s) {
            v2f a = *(const v2f*)(node + aoff + 4u * (unsigned)s);  // 8B-aligned b64 gather
            c0 = __builtin_amdgcn_wmma_f32_16x16x4_f32(false, a, false, B0[s], (short)0, c0, false, false);
            c1 = __builtin_amdgcn_wmma_f32_16x16x4_f32(false, a, false, B1[s], (short)0, c1, false, false);
        }

        // C/D layout: VGPR v, lanes 0-15 -> row M=v, N=lane; lanes 16-31 -> M=v+8.
        #pragma unroll
        for (int v = 0; v < 8; ++v) {
            int mr = v + (int)half * 8;                    // edge row within tile, 0..15
            unsigned dnode = (unsigned)__shfl((int)dst, mr, 32);  // i0 held by lane mr
            unsigned ooff  = dnode * (unsigned)OUT_C + m;  // 32-bit element offset
            unsafeAtomicAdd(out + ooff,       c0[v] + bn0);
            unsafeAtomicAdd(out + ooff + 16u, c1[v] + bn1);
        }
    }
}

extern "C" void kernel_launch(void* const* d_in, const int* in_sizes, int n_in,
                              void* d_out, int out_size, void* d_ws, size_t ws_size,
                              hipStream_t stream) {
    const float* node = (const float*)d_in[0];   // [N, 64] f32
    const int*   idx  = (const int*)d_in[1];     // [2, E]  int
    const float* Wn   = (const float*)d_in[3];   // [32, 64] f32
    const float* bn   = (const float*)d_in[4];   // [32] f32
    float* out = (float*)d_out;                  // [N, 32] f32

    int E = in_sizes[1] / 2;

    // Output must start at zero (segment_sum semantics + harness poisons d_out).
    int zb = (out_size + 255) / 256;
    gat_zero_kernel<<<zb, 256, 0, stream>>>(out, out_size);

    int numTiles = E / 16;
    if (numTiles > 0) {
        // Persistent waves: enough to saturate the chip, few enough that the
        // per-wave B preload is amortized over ~12 tiles.
        int blocks = (numTiles + 7) / 8;
        if (blocks > 1024) blocks = 1024;
        int totalWaves = blocks * 8;
        gat_edge_wmma_kernel<<<blocks, 256, 0, stream>>>(node, idx, Wn, bn, out,
                                                         E, numTiles, totalWaves);
    }
    int rem = E - numTiles * 16;
    if (rem > 0) {
        gat_edge_tail_kernel<<<1, rem * 32, 0, stream>>>(node, idx, Wn, bn, out, E, numTiles * 16);
    }
}